// MPMModel_83202106458324
// MI455X (gfx1250) — compile-verified
//
#include <hip/hip_runtime.h>
#include <math.h>

// ---------------- constants from the reference ----------------
constexpr int   NG     = 128;
constexpr float DT     = 1e-4f;
constexpr float DXC    = 1.0f / 128.0f;
constexpr float INV_DX = 128.0f;
constexpr float P_VOL  = (DXC * 0.5f) * (DXC * 0.5f);
constexpr float P_MASS = P_VOL * 1.0f;          // P_RHO = 1
constexpr float GRAV   = 9.8f;
constexpr int   GV_STRIDE = 8;                  // padded gvals stride (2x float4)

typedef float v2f __attribute__((ext_vector_type(2)));
typedef float v8f __attribute__((ext_vector_type(8)));

// D(16x16,f32) = A(16x4,f32) * B(4x16,f32) + C  -- fp32 WMMA, exact precision
__device__ __forceinline__ v8f wmma4(float a0, float a1, float b0, float b1, v8f c) {
    v2f A; A.x = a0; A.y = a1;
    v2f B; B.x = b0; B.y = b1;
    return __builtin_amdgcn_wmma_f32_16x16x4_f32(false, A, false, B, (short)0, c,
                                                 false, false);
}

// One 16x16 layer as four chained 16x16x4 WMMAs.
// W row-major [16][16]; TRANS=true uses W^T (backward pass).
// trow = this wave's LDS tile holding the input in [K][col] layout.
template <bool TRANS, bool BIAS>
__device__ __forceinline__ v8f mlp16(const float* __restrict__ W,
                                     const float* __restrict__ bias,
                                     const float (*trow)[32],
                                     int col, int lm, int koff, int mrow) {
    v8f c;
    if (BIAS) {
        const float4 blo = *(const float4*)(bias + mrow);
        const float4 bhi = *(const float4*)(bias + mrow + 4);
        c[0] = blo.x; c[1] = blo.y; c[2] = blo.z; c[3] = blo.w;
        c[4] = bhi.x; c[5] = bhi.y; c[6] = bhi.z; c[7] = bhi.w;
    } else {
#pragma unroll
        for (int r = 0; r < 8; ++r) c[r] = 0.0f;
    }
#pragma unroll
    for (int kk = 0; kk < 4; ++kk) {
        const int k0 = 4 * kk + koff;
        float a0, a1;
        if (!TRANS) {
            const float2 ap = *(const float2*)(W + lm * 16 + k0);   // 8B aligned
            a0 = ap.x; a1 = ap.y;
        } else {
            a0 = W[k0 * 16 + lm];
            a1 = W[(k0 + 1) * 16 + lm];
        }
        const float b0 = trow[k0][col];
        const float b1 = trow[k0 + 1][col];
        c = wmma4(a0, a1, b0, b1, c);
    }
    return c;
}

__global__ void mpm_zero(float* __restrict__ g, int count) {
    int i = blockIdx.x * blockDim.x + threadIdx.x;
    if (i < count) g[i] = 0.0f;
}

// ---------------- P2G + WMMA-MLP stress ----------------
__global__ __launch_bounds__(256) void mpm_p2g(
    const float* __restrict__ xin, const float* __restrict__ vin,
    const float* __restrict__ Cin, const float* __restrict__ Fin,
    const int*   __restrict__ matin, const float* __restrict__ Jpin,
    const float* __restrict__ W1, const float* __restrict__ b1,
    const float* __restrict__ W2, const float* __restrict__ b2,
    const float* __restrict__ W3, const float* __restrict__ b3,
    const float* __restrict__ W4,
    float* __restrict__ out, float* __restrict__ grid, int n)
{
    __shared__ float tile[8][16][32];     // per-wave 16x32 staging (2 WMMA tiles)

    const int tid  = threadIdx.x;
    const int wv   = tid >> 5;
    const int lane = tid & 31;
    const int lm   = lane & 15;
    const int koff = (lane >= 16) ? 2 : 0;
    const int mrow = (lane >= 16) ? 8 : 0;
    const int p    = blockIdx.x * 256 + tid;
    const bool act = (p < n);
    const int pc   = act ? p : (n - 1);
    const size_t nn = (size_t)n;
    const float (*trow)[32] = tile[wv];

    const float2 xp = *(const float2*)(xin + 2 * pc);
    const float2 vp = *(const float2*)(vin + 2 * pc);
    const float4 Cp = *(const float4*)(Cin + 4 * pc);
    const float4 Fp = *(const float4*)(Fin + 4 * pc);
    const float x0 = xp.x, x1 = xp.y, v0 = vp.x, v1 = vp.y;
    const float c00 = Cp.x, c01 = Cp.y, c10 = Cp.z, c11 = Cp.w;
    const float f00 = Fp.x, f01 = Fp.y, f10 = Fp.z, f11 = Fp.w;

    // F_new = F + DT * (C @ F)
    const float Fn00 = f00 + DT * (c00 * f00 + c01 * f10);
    const float Fn01 = f01 + DT * (c00 * f01 + c01 * f11);
    const float Fn10 = f10 + DT * (c10 * f00 + c11 * f10);
    const float Fn11 = f11 + DT * (c10 * f01 + c11 * f11);

    // Cm = Fn^T Fn (symmetric)
    const float m00 = Fn00 * Fn00 + Fn10 * Fn10;
    const float m01 = Fn00 * Fn01 + Fn10 * Fn11;
    const float m11 = Fn01 * Fn01 + Fn11 * Fn11;
    const float tr   = m00 + m11;
    const float det  = m00 * m11 - m01 * m01;
    const float gdis = tr * tr - 4.0f * det;
    const float gcl  = fmaxf(gdis, 1e-8f);
    const float delta = sqrtf(gcl);
    const float feat0 = 0.5f * (tr + delta);
    const float feat1 = 0.5f * (tr - delta);

    // stage features in B-fragment layout: rows 0,1 = feat, rows 2,3 = K-padding.
    // Each wave touches only tile[wv]; DS ops are in-order within a wave, so no
    // cross-wave barriers are needed. wave_barrier() pins scheduling order.
    tile[wv][0][lane] = feat0;
    tile[wv][1][lane] = feat1;
    tile[wv][2][lane] = 0.0f;
    tile[wv][3][lane] = 0.0f;
    __builtin_amdgcn_wave_barrier();

    v8f h1[2], h2[2], h3[2];

    // ---- layer 1: 16x2 (K padded to 4) ----
    float a10 = 0.0f, a11 = 0.0f;
    if (lane < 16) {
        const float2 w1p = *(const float2*)(W1 + lm * 2);
        a10 = w1p.x; a11 = w1p.y;
    }
    {
        const float4 blo = *(const float4*)(b1 + mrow);
        const float4 bhi = *(const float4*)(b1 + mrow + 4);
#pragma unroll
        for (int t = 0; t < 2; ++t) {
            v8f c;
            c[0] = blo.x; c[1] = blo.y; c[2] = blo.z; c[3] = blo.w;
            c[4] = bhi.x; c[5] = bhi.y; c[6] = bhi.z; c[7] = bhi.w;
            const int col = t * 16 + lm;
            c = wmma4(a10, a11, trow[koff][col], trow[koff + 1][col], c);
#pragma unroll
            for (int r = 0; r < 8; ++r) c[r] = fmaxf(c[r], 0.0f);
            h1[t] = c;
        }
    }
    __builtin_amdgcn_wave_barrier();
#pragma unroll
    for (int t = 0; t < 2; ++t)
#pragma unroll
        for (int r = 0; r < 8; ++r) tile[wv][r + mrow][t * 16 + lm] = h1[t][r];
    __builtin_amdgcn_wave_barrier();

    // ---- layer 2 ----
#pragma unroll
    for (int t = 0; t < 2; ++t) {
        v8f c = mlp16<false, true>(W2, b2, trow, t * 16 + lm, lm, koff, mrow);
#pragma unroll
        for (int r = 0; r < 8; ++r) c[r] = fmaxf(c[r], 0.0f);
        h2[t] = c;
    }
    __builtin_amdgcn_wave_barrier();
#pragma unroll
    for (int t = 0; t < 2; ++t)
#pragma unroll
        for (int r = 0; r < 8; ++r) tile[wv][r + mrow][t * 16 + lm] = h2[t][r];
    __builtin_amdgcn_wave_barrier();

    // ---- layer 3 ----
#pragma unroll
    for (int t = 0; t < 2; ++t) {
        v8f c = mlp16<false, true>(W3, b3, trow, t * 16 + lm, lm, koff, mrow);
#pragma unroll
        for (int r = 0; r < 8; ++r) c[r] = fmaxf(c[r], 0.0f);
        h3[t] = c;
    }
    __builtin_amdgcn_wave_barrier();

    // ---- backward: d3 = W4 ⊙ relu'(h3) ----
    {
        const float4 w4lo = *(const float4*)(W4 + mrow);
        const float4 w4hi = *(const float4*)(W4 + mrow + 4);
        const float w4v[8] = {w4lo.x, w4lo.y, w4lo.z, w4lo.w,
                              w4hi.x, w4hi.y, w4hi.z, w4hi.w};
#pragma unroll
        for (int t = 0; t < 2; ++t)
#pragma unroll
            for (int r = 0; r < 8; ++r)
                tile[wv][r + mrow][t * 16 + lm] = (h3[t][r] > 0.0f) ? w4v[r] : 0.0f;
    }
    __builtin_amdgcn_wave_barrier();

    v8f d2[2];
#pragma unroll
    for (int t = 0; t < 2; ++t) {
        v8f c = mlp16<true, false>(W3, nullptr, trow, t * 16 + lm, lm, koff, mrow);
#pragma unroll
        for (int r = 0; r < 8; ++r) d2[t][r] = (h2[t][r] > 0.0f) ? c[r] : 0.0f;
    }
    __builtin_amdgcn_wave_barrier();
#pragma unroll
    for (int t = 0; t < 2; ++t)
#pragma unroll
        for (int r = 0; r < 8; ++r) tile[wv][r + mrow][t * 16 + lm] = d2[t][r];
    __builtin_amdgcn_wave_barrier();

    v8f d1[2];
#pragma unroll
    for (int t = 0; t < 2; ++t) {
        v8f c = mlp16<true, false>(W2, nullptr, trow, t * 16 + lm, lm, koff, mrow);
#pragma unroll
        for (int r = 0; r < 8; ++r) d1[t][r] = (h1[t][r] > 0.0f) ? c[r] : 0.0f;
    }
    __builtin_amdgcn_wave_barrier();
#pragma unroll
    for (int t = 0; t < 2; ++t)
#pragma unroll
        for (int r = 0; r < 8; ++r) tile[wv][r + mrow][t * 16 + lm] = d1[t][r];
    __builtin_amdgcn_wave_barrier();

    // dfeat = W1^T @ d1 (per-particle column = this lane)
    float gf0 = 0.0f, gf1 = 0.0f;
#pragma unroll
    for (int m = 0; m < 16; ++m) {
        const float dm = trow[m][lane];
        const float2 w1p = *(const float2*)(W1 + m * 2);
        gf0 += w1p.x * dm;
        gf1 += w1p.y * dm;
    }

    // chain rule through eigenvalue features back to F
    const float ddelta  = 0.5f * (gf0 - gf1);
    const float dtr_dir = 0.5f * (gf0 + gf1);
    const float dg   = (gdis >= 1e-8f) ? (ddelta / (2.0f * delta)) : 0.0f;
    const float dtr  = dtr_dir + dg * 2.0f * tr;
    const float ddet = -4.0f * dg;
    const float G00  = dtr + ddet * m11;
    const float G11  = dtr + ddet * m00;
    const float Goff = -ddet * m01;
    const float S00 = 2.0f * G00, S11 = 2.0f * G11, S01 = 2.0f * Goff;
    const float dF00 = Fn00 * S00 + Fn01 * S01;
    const float dF01 = Fn00 * S01 + Fn01 * S11;
    const float dF10 = Fn10 * S00 + Fn11 * S01;
    const float dF11 = Fn10 * S01 + Fn11 * S11;

    const float sc  = -DT * P_VOL * 4.0f * INV_DX * INV_DX;
    const float a00 = sc * dF00 + P_MASS * c00;
    const float a01 = sc * dF01 + P_MASS * c01;
    const float aa10 = sc * dF10 + P_MASS * c10;
    const float aa11 = sc * dF11 + P_MASS * c11;
    const float va0 = P_MASS * v0 - (a00 * x0 + a01 * x1);
    const float va1 = P_MASS * v1 - (aa10 * x0 + aa11 * x1);

    // quadratic B-spline weights + scatter
    const float px = x0 * INV_DX, py = x1 * INV_DX;
    const int bi = (int)floorf(px - 0.5f);
    const int bj = (int)floorf(py - 0.5f);
    const float fx0 = px - (float)bi, fx1 = py - (float)bj;
    float wx[3], wy[3];
    wx[0] = 0.5f * (1.5f - fx0) * (1.5f - fx0);
    wx[1] = 0.75f - (fx0 - 1.0f) * (fx0 - 1.0f);
    wx[2] = 0.5f * (fx0 - 0.5f) * (fx0 - 0.5f);
    wy[0] = 0.5f * (1.5f - fx1) * (1.5f - fx1);
    wy[1] = 0.75f - (fx1 - 1.0f) * (fx1 - 1.0f);
    wy[2] = 0.5f * (fx1 - 0.5f) * (fx1 - 0.5f);

    if (act) {
        const float vals[7] = {va0, va1, a00, a01, aa10, aa11, P_MASS};
#pragma unroll
        for (int i = 0; i < 3; ++i)
#pragma unroll
            for (int j = 0; j < 3; ++j) {
                const int gi = bi + i, gj = bj + j;
                if (gi >= 0 && gi < NG && gj >= 0 && gj < NG) {
                    const float wt = wx[i] * wy[j];
                    float* cell = grid + (size_t)(gi * NG + gj) * 7;
#pragma unroll
                    for (int ch = 0; ch < 7; ++ch) atomicAdd(cell + ch, wt * vals[ch]);
                }
            }
        // outputs: x_new, F_new, material (bit copy), Jp
        *(float2*)(out + 2 * p) = make_float2(x0 + DT * v0, x1 + DT * v1);
        *(float4*)(out + 8 * nn + 4 * p) = make_float4(Fn00, Fn01, Fn10, Fn11);
        out[12 * nn + p] = __int_as_float(matin[pc]);
        out[13 * nn + p] = Jpin[pc];
    }
}

// ---------------- grid update ----------------
__global__ void mpm_grid(const float* __restrict__ grid, float* __restrict__ gvals) {
    const int cell = blockIdx.x * blockDim.x + threadIdx.x;
    if (cell >= NG * NG) return;
    const int i = cell / NG, j = cell % NG;
    const float* g = grid + (size_t)cell * 7;
    float gv0 = g[0], gv1 = g[1];
    const float a00 = g[2], a01 = g[3], a10 = g[4], a11 = g[5], m = g[6];
    const float gx0 = i * DXC, gx1 = j * DXC;
    gv0 += a00 * gx0 + a01 * gx1;
    gv1 += a10 * gx0 + a11 * gx1;
    if (m > 0.0f) { gv0 /= m; gv1 /= m; }
    gv1 -= DT * GRAV;
    if (i < 3)       gv0 = fmaxf(gv0, 0.0f);
    if (i >= NG - 3) gv0 = fminf(gv0, 0.0f);
    if (j < 3)       gv1 = fmaxf(gv1, 0.0f);
    if (j >= NG - 3) gv1 = fminf(gv1, 0.0f);
    float* o = gvals + (size_t)cell * GV_STRIDE;
    *(float4*)(o)     = make_float4(gv0, gv1, gv0 * gx0, gv0 * gx1);
    *(float4*)(o + 4) = make_float4(gv1 * gx0, gv1 * gx1, 0.0f, 0.0f);
}

// ---------------- G2P gather ----------------
__global__ void mpm_g2p(const float* __restrict__ xin, float* __restrict__ out,
                        const float* __restrict__ gvals, int n) {
    const int p = blockIdx.x * blockDim.x + threadIdx.x;
    if (p >= n) return;
    const size_t nn = (size_t)n;
    const float2 xp = *(const float2*)(xin + 2 * p);
    const float x0 = xp.x, x1 = xp.y;
    const float px = x0 * INV_DX, py = x1 * INV_DX;
    const int bi = (int)floorf(px - 0.5f);
    const int bj = (int)floorf(py - 0.5f);
    const float fx0 = px - (float)bi, fx1 = py - (float)bj;
    float wx[3], wy[3];
    wx[0] = 0.5f * (1.5f - fx0) * (1.5f - fx0);
    wx[1] = 0.75f - (fx0 - 1.0f) * (fx0 - 1.0f);
    wx[2] = 0.5f * (fx0 - 0.5f) * (fx0 - 0.5f);
    wy[0] = 0.5f * (1.5f - fx1) * (1.5f - fx1);
    wy[1] = 0.75f - (fx1 - 1.0f) * (fx1 - 1.0f);
    wy[2] = 0.5f * (fx1 - 0.5f) * (fx1 - 0.5f);
    float acc[6] = {0, 0, 0, 0, 0, 0};
#pragma unroll
    for (int i = 0; i < 3; ++i)
#pragma unroll
        for (int j = 0; j < 3; ++j) {
            const int gi = bi + i, gj = bj + j;
            if (gi >= 0 && gi < NG && gj >= 0 && gj < NG) {
                const float wt = wx[i] * wy[j];
                const float* gv = gvals + (size_t)(gi * NG + gj) * GV_STRIDE;
                const float4 ga = *(const float4*)(gv);
                const float4 gb = *(const float4*)(gv + 4);
                acc[0] += wt * ga.x; acc[1] += wt * ga.y;
                acc[2] += wt * ga.z; acc[3] += wt * ga.w;
                acc[4] += wt * gb.x; acc[5] += wt * gb.y;
            }
        }
    const float nv0 = acc[0], nv1 = acc[1];
    const float k4 = 4.0f * INV_DX * INV_DX;
    *(float2*)(out + 2 * nn + 2 * p) = make_float2(nv0, nv1);
    *(float4*)(out + 4 * nn + 4 * p) =
        make_float4(k4 * (acc[2] - nv0 * x0), k4 * (acc[3] - nv0 * x1),
                    k4 * (acc[4] - nv1 * x0), k4 * (acc[5] - nv1 * x1));
}

extern "C" void kernel_launch(void* const* d_in, const int* in_sizes, int n_in,
                              void* d_out, int out_size, void* d_ws, size_t ws_size,
                              hipStream_t stream) {
    (void)n_in; (void)out_size; (void)ws_size;
    const float* x   = (const float*)d_in[0];
    const float* v   = (const float*)d_in[1];
    const float* C   = (const float*)d_in[2];
    const float* F   = (const float*)d_in[3];
    const int*   mat = (const int*)d_in[4];
    const float* Jp  = (const float*)d_in[5];
    // d_in[6]=E, d_in[7]=nu unused by the reference math path
    const float* W1 = (const float*)d_in[8];
    const float* b1 = (const float*)d_in[9];
    const float* W2 = (const float*)d_in[10];
    const float* b2 = (const float*)d_in[11];
    const float* W3 = (const float*)d_in[12];
    const float* b3 = (const float*)d_in[13];
    const float* W4 = (const float*)d_in[14];

    const int n = in_sizes[0] / 2;
    float* gridAcc = (float*)d_ws;                      // NG*NG*7 floats
    float* gvals   = gridAcc + NG * NG * 7;             // NG*NG*GV_STRIDE floats

    const int gcnt = NG * NG * 7;
    mpm_zero<<<(gcnt + 255) / 256, 256, 0, stream>>>(gridAcc, gcnt);

    const int pblocks = (n + 255) / 256;
    mpm_p2g<<<pblocks, 256, 0, stream>>>(x, v, C, F, mat, Jp,
                                         W1, b1, W2, b2, W3, b3, W4,
                                         (float*)d_out, gridAcc, n);
    mpm_grid<<<(NG * NG + 255) / 256, 256, 0, stream>>>(gridAcc, gvals);
    mpm_g2p<<<pblocks, 256, 0, stream>>>(x, (float*)d_out, gvals, n);
}